// V7_loss_54503134986572
// MI455X (gfx1250) — compile-verified
//
#include <hip/hip_runtime.h>
#include <stdint.h>

// ---------------- problem constants ----------------
#define NB    256      // batches
#define TT    1024     // time steps
#define FF    159      // features
#define FP    160      // block size (5 wave32s)
#define CH    128      // anchor rows per chunk
#define NCH   (TT/CH)  // 8 chunks per batch
#define TS    16       // time-rows per LDS tile (one TDM descriptor per tensor)
#define HALO  7        // derivative order halo
#define NLVL  7
#define EPSF  1e-10f

typedef unsigned int u32x4 __attribute__((ext_vector_type(4)));
typedef int          i32x4 __attribute__((ext_vector_type(4)));
typedef int          i32x8 __attribute__((ext_vector_type(8)));

// One TDM descriptor: 2D tile, rows x FF fp32, contiguous rows (stride FF),
// global -> LDS. Tracked with TENSORcnt. See CDNA5 ISA.
__device__ __forceinline__ void tdm_load_tile(uint32_t lds_byte,
                                              const float* gptr, int rows) {
    const uint64_t ga = (uint64_t)(uintptr_t)gptr;
    u32x4 g0;
    g0[0] = 1u;                                   // count=1, no gather
    g0[1] = lds_byte;                             // lds_addr [63:32]
    g0[2] = (uint32_t)ga;                         // global_addr [95:64]
    g0[3] = (uint32_t)((ga >> 32) & 0x1FFFFFFu)   // global_addr [120:96]
          | (2u << 30);                           // type=2 ("image") [127:126]
    i32x8 g1;
    g1[0] = 0x20000;                              // data_size=2 (4 bytes)
    g1[1] = (int)((unsigned)FF << 16);            // tensor_dim0[15:0] @ [63:48]
    g1[2] = (int)((unsigned)rows << 16);          // tensor_dim1[15:0] @ [95:80]
    g1[3] = (int)((unsigned)FF << 16);            // tile_dim0 @ [127:112]
    g1[4] = rows;                                 // tile_dim1 @ [143:128], tile_dim2=0
    g1[5] = FF;                                   // tensor_dim0_stride[31:0]
    g1[6] = 0;                                    // stride0 hi / stride1 lo
    g1[7] = 0;                                    // stride1 hi
    i32x4 z4 = {0, 0, 0, 0};
#if defined(__clang_major__) && (__clang_major__ >= 23)
    i32x8 z8 = {0, 0, 0, 0, 0, 0, 0, 0};
    __builtin_amdgcn_tensor_load_to_lds(g0, g1, z4, z4, z8, 0);
#else
    __builtin_amdgcn_tensor_load_to_lds(g0, g1, z4, z4, 0);
#endif
}

__global__ __launch_bounds__(FP)
void v7_partial(const float* __restrict__ pred,
                const float* __restrict__ gt,
                float* __restrict__ ws) {
    __shared__ __align__(16) float sp[2][TS * FF];   // pred tiles (double buffer)
    __shared__ __align__(16) float sg[2][TS * FF];   // gt tiles
    __shared__ float sred[NLVL][FP];                 // block reduction scratch

    const int f     = threadIdx.x;
    const int b     = blockIdx.x / NCH;
    const int chunk = blockIdx.x % NCH;
    const int T0    = chunk * CH;                     // first anchor row
    const int R0    = (T0 >= HALO) ? (T0 - HALO) : 0; // first streamed row
    const int T1    = T0 + CH;                        // exclusive end
    const int nrows = T1 - R0;                        // 128 or 135
    const int ntiles = (nrows + TS - 1) / TS;

    const float* pbase = pred + (size_t)b * TT * FF;
    const float* gbase = gt   + (size_t)b * TT * FF;

    // TDM ignores EXEC -> must branch around it at *wave* granularity with a
    // scalar condition, or every wave of the block would issue duplicate DMAs.
    const int waveid = __builtin_amdgcn_readfirstlane((int)(threadIdx.x >> 5));

    auto issue = [&](int tile, int buf) {   // wave 0 only; 2 TDM descriptors
        const int r0   = R0 + tile * TS;
        const int rend = (r0 + TS < T1) ? (r0 + TS) : T1;
        const int rows = rend - r0;
        tdm_load_tile((uint32_t)(uintptr_t)&sp[buf][0], pbase + (size_t)r0 * FF, rows);
        tdm_load_tile((uint32_t)(uintptr_t)&sg[buf][0], gbase + (size_t)r0 * FF, rows);
    };

    float pprev[NLVL], gprev[NLVL], acc[NLVL];
#pragma unroll
    for (int k = 0; k < NLVL; ++k) acc[k] = 0.0f;

    if (waveid == 0) issue(0, 0);

    for (int tile = 0; tile < ntiles; ++tile) {
        const int buf = tile & 1;
        if (waveid == 0) {
            if (tile + 1 < ntiles) {
                issue(tile + 1, buf ^ 1);                 // prefetch next tile
                __builtin_amdgcn_s_wait_tensorcnt(2);     // this tile landed
            } else {
                __builtin_amdgcn_s_wait_tensorcnt(0);
            }
        }
        __syncthreads();

        const int r0   = R0 + tile * TS;
        const int rend = (r0 + TS < T1) ? (r0 + TS) : T1;
        if (f < FF) {
            if (tile == 0) {
                // warm-up: uniform-branchy cascade bootstrap (<= TS rows total)
                for (int r = r0; r < rend; ++r) {
                    const int  s     = r - R0;
                    const bool doAcc = (r >= T0);
                    float pc = sp[buf][(r - r0) * FF + f];
                    float gc = sg[buf][(r - r0) * FF + f];
#pragma unroll
                    for (int k = 0; k < NLVL; ++k) {
                        if (s > k) {
                            const float pn = fabsf(pc - pprev[k] + EPSF);
                            const float gn = fabsf(gc - gprev[k] + EPSF);
                            pprev[k] = pc; gprev[k] = gc;
                            pc = pn; gc = gn;
                            if (doAcc) {
                                const float d = 10.0f * fabsf(pc - gc);
                                const float m = fminf(d, 1.0f);
                                acc[k] += fmaf(0.5f * m, m, d - m);
                            }
                        } else if (s == k) {
                            pprev[k] = pc; gprev[k] = gc;
                        }
                    }
                }
            } else {
                // steady state: s >= TS > 7 for every row -> all levels live,
                // r > T0 always -> always accumulate. Pure VALU + 2 ds loads.
                for (int r = r0; r < rend; ++r) {
                    float pc = sp[buf][(r - r0) * FF + f];
                    float gc = sg[buf][(r - r0) * FF + f];
#pragma unroll
                    for (int k = 0; k < NLVL; ++k) {
                        const float pn = fabsf(pc - pprev[k] + EPSF);
                        const float gn = fabsf(gc - gprev[k] + EPSF);
                        pprev[k] = pc; gprev[k] = gc;
                        pc = pn; gc = gn;
                        const float d = 10.0f * fabsf(pc - gc);   // |p-g|/beta
                        const float m = fminf(d, 1.0f);
                        acc[k] += fmaf(0.5f * m, m, d - m);       // smooth-L1
                    }
                }
            }
        }
        __syncthreads();   // buffer free for reuse next iteration
    }

    // deterministic block reduction
#pragma unroll
    for (int k = 0; k < NLVL; ++k) sred[k][f] = (f < FF) ? acc[k] : 0.0f;
    __syncthreads();
    if (f < NLVL) {
        float ssum = 0.0f;
        for (int i = 0; i < FP; ++i) ssum += sred[f][i];
        ws[(size_t)blockIdx.x * NLVL + f] = ssum;
    }
}

__global__ void v7_final(const float* __restrict__ ws,
                         float* __restrict__ out, int nblocks) {
    const int k = threadIdx.x;
    if (k < NLVL) {
        double dsum = 0.0;                        // fixed order, deterministic
        for (int i = 0; i < nblocks; ++i) dsum += (double)ws[(size_t)i * NLVL + k];
        const double n = (double)NB * (double)(TT - (k + 1)) * (double)FF;
        out[k] = (float)(dsum / n * 0.1);         // mean * beta
    }
}

extern "C" void kernel_launch(void* const* d_in, const int* in_sizes, int n_in,
                              void* d_out, int out_size, void* d_ws, size_t ws_size,
                              hipStream_t stream) {
    const float* pred = (const float*)d_in[0];
    const float* gt   = (const float*)d_in[1];
    float* out = (float*)d_out;
    float* ws  = (float*)d_ws;      // needs 2048*7*4 = 57344 bytes

    const int nblocks = NB * NCH;   // 2048
    v7_partial<<<nblocks, FP, 0, stream>>>(pred, gt, ws);
    v7_final<<<1, 32, 0, stream>>>(ws, out, nblocks);
}